// Encoder_26336739459290
// MI455X (gfx1250) — compile-verified
//
#include <hip/hip_runtime.h>
#include <hip/hip_bf16.h>

typedef __attribute__((ext_vector_type(2))) float v2f;
typedef __attribute__((ext_vector_type(8))) float v8f;
typedef int v4i __attribute__((vector_size(16)));  // matches builtin param type

#define TPB 256
#define BROWS 128     // rows per block (8 waves x 16)
#define BCOLS 64      // cols per block (4 wmma col tiles per wave)
#define KT 32         // K-tile (8 wmma k-steps)
#define SA_STRIDE 36  // padded A row stride (floats): float4-aligned, bank-clean
#define SB_STRIDE 160 // padded B pair-row stride (floats): half-waves hit disjoint banks

#define AS1 __attribute__((address_space(1)))
#define AS3 __attribute__((address_space(3)))

#if defined(__has_builtin)
#if __has_builtin(__builtin_amdgcn_global_load_async_to_lds_b128)
#define HAVE_ASYNC_LDS 1
#endif
#endif

__device__ __forceinline__ void wait_async0() {
#if defined(__has_builtin) && __has_builtin(__builtin_amdgcn_s_wait_asynccnt)
  __builtin_amdgcn_s_wait_asynccnt(0);
#else
  asm volatile("s_wait_asynccnt 0x0" ::: "memory");
#endif
}

// ---------------- GEMM: C[N,M] = A[N,K] * B[K,M], fp32 via V_WMMA_F32_16X16X4_F32 ----
// Double-buffered LDS; A tile staged with async global->LDS (ASYNCcnt), B tile staged
// through registers into a k-pair-interleaved layout (1x ds b64 per fragment).
__global__ __launch_bounds__(TPB) void gemm_wmma_f32(
    const float* __restrict__ A, const float* __restrict__ B,
    float* __restrict__ C, int N, int K, int M) {
  __shared__ float sA[2][BROWS * SA_STRIDE];     // 2 x 18432 B
  __shared__ float sB[2][(KT / 2) * SB_STRIDE];  // 2 x 10240 B
  const int tid  = threadIdx.x;
  const int lane = tid & 31;
  const int wave = tid >> 5;      // 0..7
  const int lrow = lane & 15;
  const int lhi  = lane >> 4;     // 0/1 half-wave
  const int rowBase = blockIdx.x * BROWS;
  const int colBase = blockIdx.y * BCOLS;

  const int r0 = tid >> 3;           // A staging: row 0..31
  const int cg = (tid & 7) << 2;     // A staging: k-col 0,4,...,28
  const int kb = tid >> 4;           // B staging: k 0..15
  const int mg = (tid & 15) << 2;    // B staging: col 0,4,...,60

  v8f acc[4];
#pragma unroll
  for (int j = 0; j < 4; ++j)
#pragma unroll
    for (int i = 0; i < 8; ++i) acc[j][i] = 0.0f;

  auto stage = [&](int kt, int buf) {
    // ---- A tile (BROWS x KT), row-clamped ----
#ifdef HAVE_ASYNC_LDS
#pragma unroll
    for (int p = 0; p < 4; ++p) {
      const int r = r0 + p * 32;
      int gr = rowBase + r;
      gr = (gr < N) ? gr : (N - 1);
      __builtin_amdgcn_global_load_async_to_lds_b128(
          (AS1 v4i*)(A + (size_t)gr * K + kt + cg),
          (AS3 v4i*)(&sA[buf][r * SA_STRIDE + cg]), 0, 0);
    }
#else
    float4 av[4];
#pragma unroll
    for (int p = 0; p < 4; ++p) {
      int gr = rowBase + r0 + p * 32;
      gr = (gr < N) ? gr : (N - 1);
      av[p] = *(const float4*)(A + (size_t)gr * K + kt + cg);
    }
#pragma unroll
    for (int p = 0; p < 4; ++p)
      *(float4*)(&sA[buf][(r0 + p * 32) * SA_STRIDE + cg]) = av[p];
#endif
    // ---- B tile (KT x BCOLS), pair-interleaved:
    // sB[(k/2)*SB_STRIDE + col*2 + (k&1)] = B[kt+k][colBase+col]
    float4 bv[2];
#pragma unroll
    for (int p = 0; p < 2; ++p)
      bv[p] = *(const float4*)(B + (size_t)(kt + kb + p * 16) * M + colBase + mg);
#pragma unroll
    for (int p = 0; p < 2; ++p) {
      const int k = kb + p * 16;
      float* d = &sB[buf][(k >> 1) * SB_STRIDE + (k & 1)];
      d[(mg + 0) * 2] = bv[p].x;
      d[(mg + 1) * 2] = bv[p].y;
      d[(mg + 2) * 2] = bv[p].z;
      d[(mg + 3) * 2] = bv[p].w;
    }
  };

  stage(0, 0);
  int buf = 0;
  for (int kt = 0; kt < K; kt += KT) {
#ifdef HAVE_ASYNC_LDS
    wait_async0();            // our async A chunks landed in LDS
#endif
    __syncthreads();          // everyone's staging visible; prev-tile reads done
    if (kt + KT < K) stage(kt + KT, buf ^ 1);

    const float* sAb = &sA[buf][0];
    const float* sBb = &sB[buf][0];
    // preload all 8 A fragments of this K-tile
    v2f af[8];
#pragma unroll
    for (int s = 0; s < 8; ++s)
      af[s] = *(const v2f*)(sAb + (wave * 16 + lrow) * SA_STRIDE + s * 4 + lhi * 2);

    // software-pipelined B fragments: issue step s+1 before WMMAs of step s
    v2f bc[4], bn[4];
#pragma unroll
    for (int j = 0; j < 4; ++j)
      bc[j] = *(const v2f*)(sBb + (0 * 2 + lhi) * SB_STRIDE + ((j * 16 + lrow) << 1));
#pragma unroll
    for (int s = 0; s < 8; ++s) {
      if (s < 7) {
#pragma unroll
        for (int j = 0; j < 4; ++j)
          bn[j] = *(const v2f*)(sBb + ((s + 1) * 2 + lhi) * SB_STRIDE + ((j * 16 + lrow) << 1));
      }
#pragma unroll
      for (int j = 0; j < 4; ++j)
        acc[j] = __builtin_amdgcn_wmma_f32_16x16x4_f32(
            false, af[s], false, bc[j], (short)0, acc[j], false, false);
      if (s < 7) {
#pragma unroll
        for (int j = 0; j < 4; ++j) bc[j] = bn[j];
      }
    }
    buf ^= 1;
  }

  // D layout: VGPR i -> M = i + 8*halfwave; N = lane%16
  const int rw = rowBase + wave * 16 + lhi * 8;
#pragma unroll
  for (int j = 0; j < 4; ++j) {
    const int col = colBase + j * 16 + lrow;
#pragma unroll
    for (int i = 0; i < 8; ++i) {
      const int row = rw + i;
      if (row < N) C[(size_t)row * M + col] = acc[j][i];
    }
  }
}

// ---------------- graph helpers ----------------
__global__ void fill_zero_kernel(float* __restrict__ p, long long n4) {
  const long long i = (long long)blockIdx.x * blockDim.x + threadIdx.x;
  if (i < n4) ((float4*)p)[i] = make_float4(0.f, 0.f, 0.f, 0.f);
}

__global__ void init_deg_kernel(float* __restrict__ deg, int N) {
  const int i = blockIdx.x * blockDim.x + threadIdx.x;
  if (i < N) deg[i] = 1.0f;  // self-loop contributes 1
}

__global__ void count_deg_kernel(const int* __restrict__ dst, float* __restrict__ deg, int E) {
  const int e = blockIdx.x * blockDim.x + threadIdx.x;
  if (e < E) atomicAdd(&deg[dst[e]], 1.0f);
}

__global__ void rsqrt_kernel(float* __restrict__ deg, int N) {
  const int i = blockIdx.x * blockDim.x + threadIdx.x;
  if (i < N) deg[i] = rsqrtf(deg[i]);  // deg >= 1 always (self loops)
}

// out[dst] += xw[src] * dis[src]*dis[dst], (F/4) threads per edge, float4 gathers
__global__ void scatter_kernel(const float* __restrict__ xw, const float* __restrict__ dis,
                               const int* __restrict__ src, const int* __restrict__ dst,
                               float* __restrict__ agg, int E, int F, int shift) {
  const long long gid = (long long)blockIdx.x * blockDim.x + threadIdx.x;
  const int e = (int)(gid >> shift);
  if (e >= E) return;
  const int f4 = ((int)gid & ((1 << shift) - 1)) << 2;
  const int s = src[e];
  const int d = dst[e];
  const float nrm = dis[s] * dis[d];
  const float4 v = *(const float4*)(xw + (size_t)s * F + f4);
  float* o = agg + (size_t)d * F + f4;
  atomicAdd(o + 0, v.x * nrm);
  atomicAdd(o + 1, v.y * nrm);
  atomicAdd(o + 2, v.z * nrm);
  atomicAdd(o + 3, v.w * nrm);
}

// agg = (relu?)(agg + xw*dis^2 + bias)  -- fuses self-loop message, bias, activation
__global__ void epilogue_kernel(float* __restrict__ agg, const float* __restrict__ xw,
                                const float* __restrict__ dis, const float* __restrict__ bias,
                                int N, int F, int shift, int relu) {
  const long long gid = (long long)blockIdx.x * blockDim.x + threadIdx.x;
  const int node = (int)(gid >> shift);
  if (node >= N) return;
  const int f4 = ((int)gid & ((1 << shift) - 1)) << 2;
  const float dd = dis[node];
  const float nrm = dd * dd;
  float4 v = ((float4*)agg)[gid];
  const float4 w = ((const float4*)xw)[gid];
  v.x += w.x * nrm + bias[f4 + 0];
  v.y += w.y * nrm + bias[f4 + 1];
  v.z += w.z * nrm + bias[f4 + 2];
  v.w += w.w * nrm + bias[f4 + 3];
  if (relu) {
    v.x = fmaxf(v.x, 0.f);
    v.y = fmaxf(v.y, 0.f);
    v.z = fmaxf(v.z, 0.f);
    v.w = fmaxf(v.w, 0.f);
  }
  ((float4*)agg)[gid] = v;
}

static inline unsigned cdiv(long long a, long long b) { return (unsigned)((a + b - 1) / b); }

extern "C" void kernel_launch(void* const* d_in, const int* in_sizes, int n_in,
                              void* d_out, int out_size, void* d_ws, size_t ws_size,
                              hipStream_t stream) {
  const float* x  = (const float*)d_in[0];
  const int*   ei = (const int*)d_in[1];
  const float* W1 = (const float*)d_in[2];
  const float* b1 = (const float*)d_in[3];
  const float* W2 = (const float*)d_in[4];
  const float* b2 = (const float*)d_in[5];

  const int HID  = in_sizes[3];            // 256
  const int OUT  = in_sizes[5];            // 128
  const int INCH = in_sizes[2] / HID;      // 512
  const int N    = in_sizes[0] / INCH;     // 50000
  const int E    = in_sizes[1] / 2;        // 800000
  const int* src = ei;
  const int* dst = ei + E;
  float* out = (float*)d_out;

  // workspace layout
  float* dis  = (float*)d_ws;                              // N (deg -> deg^-1/2)
  float* xw1  = dis + (((size_t)N + 255) & ~(size_t)255);  // N*HID
  float* agg1 = xw1 + (size_t)N * HID;                     // N*HID (becomes h after epilogue)
  float* hw2  = agg1 + (size_t)N * HID;                    // N*OUT

  // --- degree / normalization ---
  init_deg_kernel<<<cdiv(N, 256), 256, 0, stream>>>(dis, N);
  count_deg_kernel<<<cdiv(E, 256), 256, 0, stream>>>(dst, dis, E);
  rsqrt_kernel<<<cdiv(N, 256), 256, 0, stream>>>(dis, N);

  // --- layer 1: xw1 = x @ W1 ; agg1 = scatter + selfloop + b1, ReLU ---
  gemm_wmma_f32<<<dim3(cdiv(N, BROWS), HID / BCOLS), TPB, 0, stream>>>(x, W1, xw1, N, INCH, HID);
  fill_zero_kernel<<<cdiv((long long)N * HID / 4, 256), 256, 0, stream>>>(agg1, (long long)N * HID / 4);
  const int sh1 = __builtin_ctz(HID >> 2);  // log2(F/4) = 6
  scatter_kernel<<<cdiv((long long)E << sh1, 256), 256, 0, stream>>>(xw1, dis, src, dst, agg1, E, HID, sh1);
  epilogue_kernel<<<cdiv((long long)N << sh1, 256), 256, 0, stream>>>(agg1, xw1, dis, b1, N, HID, sh1, 1);

  // --- layer 2: hw2 = h @ W2 ; out = scatter + selfloop + b2 ---
  gemm_wmma_f32<<<dim3(cdiv(N, BROWS), OUT / BCOLS), TPB, 0, stream>>>(agg1, W2, hw2, N, HID, OUT);
  const int sh2 = __builtin_ctz(OUT >> 2);  // 5
  fill_zero_kernel<<<cdiv((long long)N * OUT / 4, 256), 256, 0, stream>>>(out, (long long)N * OUT / 4);
  scatter_kernel<<<cdiv((long long)E << sh2, 256), 256, 0, stream>>>(hw2, dis, src, dst, out, E, OUT, sh2);
  epilogue_kernel<<<cdiv((long long)N << sh2, 256), 256, 0, stream>>>(out, hw2, dis, b2, N, OUT, sh2, 0);
}